// AdditiveAttention_74217034875299
// MI455X (gfx1250) — compile-verified
//
#include <hip/hip_runtime.h>

typedef __attribute__((ext_vector_type(2))) float v2f;
typedef __attribute__((ext_vector_type(8))) float v8f;

#if __has_builtin(__builtin_amdgcn_tanhf)
__device__ __forceinline__ float fast_tanh(float x) { return __builtin_amdgcn_tanhf(x); }
#elif __has_builtin(__builtin_amdgcn_tanh_f32)
__device__ __forceinline__ float fast_tanh(float x) { return __builtin_amdgcn_tanh_f32(x); }
#else
__device__ __forceinline__ float fast_tanh(float x) { return tanhf(x); }
#endif

#define BB   32
#define QN   256
#define KN   256
#define HH   128
#define QT   16     // queries per workgroup
#define KC   64     // key chunk staged in LDS
#define KPAD 132    // 128 + 4  -> bank stride 4, float4-store aligned
#define SPAD 260    // 256 + 4  -> bank stride 4 for WMMA A reads

__global__ __launch_bounds__(256, 2)
void addattn_kernel(const float* __restrict__ qg,
                    const float* __restrict__ kg,
                    const float* __restrict__ vg,
                    const float* __restrict__ wg,
                    float* __restrict__ out)
{
    __shared__ float sQ[QT][KPAD];   // 8.25 KB
    __shared__ float sK[KC][KPAD];   // 33 KB
    __shared__ float sS[QT][SPAD];   // 16.25 KB  (scores -> attn in place)
    __shared__ float sW[HH];

    const int tid = threadIdx.x;
    const int b   = blockIdx.x >> 4;          // Q/QT == 16 tiles per batch
    const int q0  = (blockIdx.x & 15) * QT;

    if (tid < HH) sW[tid] = wg[tid];

    // ---- stage Q tile: 16x128 f32 = 512 float4 ----
    {
        const float4* src = (const float4*)(qg + (size_t)(b * QN + q0) * HH);
        for (int i = tid; i < QT * HH / 4; i += 256) {
            int row = i >> 5;                 // 32 float4 per row
            int c4  = i & 31;
            float4 v = src[row * (HH / 4) + c4];
            sQ[row][c4 * 4 + 0] = v.x; sQ[row][c4 * 4 + 1] = v.y;
            sQ[row][c4 * 4 + 2] = v.z; sQ[row][c4 * 4 + 3] = v.w;
        }
    }

    const int qi = tid & 15;                  // score-phase q row (16 lanes share a k row)
    for (int kc = 0; kc < KN; kc += KC) {
        __syncthreads();
        // ---- stage K chunk: 64x128 f32 = 2048 float4 ----
        const float4* src = (const float4*)(kg + (size_t)(b * KN + kc) * HH);
        for (int i = tid; i < KC * HH / 4; i += 256) {
            int row = i >> 5;
            int c4  = i & 31;
            float4 v = src[row * (HH / 4) + c4];
            sK[row][c4 * 4 + 0] = v.x; sK[row][c4 * 4 + 1] = v.y;
            sK[row][c4 * 4 + 2] = v.z; sK[row][c4 * 4 + 3] = v.w;
        }
        __syncthreads();
        // ---- scores: 16q x 64k pairs, 4 pairs/thread, 128-deep tanh-FMA ----
        for (int i = 0; i < 4; ++i) {
            const int ki = i * 16 + (tid >> 4);
            const float* __restrict__ qr = sQ[qi];
            const float* __restrict__ kr = sK[ki];
            float s = 0.f;
            #pragma unroll 4
            for (int h = 0; h < HH; ++h)
                s = fmaf(fast_tanh(qr[h] + kr[h]), sW[h], s);
            sS[qi][kc + ki] = s;
        }
    }
    __syncthreads();

    // ---- softmax over k (rows of 256), one wave per 2 rows ----
    const int lane = tid & 31;
    const int wave = tid >> 5;
    for (int r = wave; r < QT; r += 8) {
        float ev[KN / 32];
        float m = -3.4e38f;
        for (int j = 0; j < KN / 32; ++j) m = fmaxf(m, sS[r][lane + 32 * j]);
        for (int off = 16; off > 0; off >>= 1) m = fmaxf(m, __shfl_xor(m, off, 32));
        float sum = 0.f;
        for (int j = 0; j < KN / 32; ++j) {
            ev[j] = __expf(sS[r][lane + 32 * j] - m);
            sum += ev[j];
        }
        for (int off = 16; off > 0; off >>= 1) sum += __shfl_xor(sum, off, 32);
        const float inv = 1.0f / sum;
        for (int j = 0; j < KN / 32; ++j) sS[r][lane + 32 * j] = ev[j] * inv;
    }
    __syncthreads();

    // ---- PV GEMM: attn[16x256] @ V[256x128] via V_WMMA_F32_16X16X4_F32 ----
    // wave w owns output columns [16w, 16w+16); 64 K-steps of 4.
    const int h0   = wave * 16;
    const int half = lane >> 4;               // K-pair selector per ISA A-layout
    const int l    = lane & 15;               // M for A, N for B/D
    v8f acc = {0.f, 0.f, 0.f, 0.f, 0.f, 0.f, 0.f, 0.f};
    const float* __restrict__ vb = vg + (size_t)b * KN * HH + h0 + l;
    for (int k = 0; k < KN; k += 4) {
        v2f a, bf;
        // A 16x4: lanes0-15 {K=k,k+1}, lanes16-31 {K=k+2,k+3} (ISA 7.12.2)
        a[0] = sS[l][k + 2 * half];
        a[1] = sS[l][k + 2 * half + 1];
        // B 4x16: symmetric K split across lane halves, N striped on lanes
        const float* vp = vb + (size_t)(k + 2 * half) * HH;
        bf[0] = vp[0];
        bf[1] = vp[HH];
        acc = __builtin_amdgcn_wmma_f32_16x16x4_f32(
            /*neg_a=*/false, a, /*neg_b=*/false, bf,
            /*c_mod=*/(short)0, acc, /*reuse_a=*/false, /*reuse_b=*/false);
    }
    // D 16x16 f32: VGPR r -> M=r (lanes0-15) / M=r+8 (lanes16-31), N=l
    float* ob = out + (size_t)(b * QN + q0) * HH + h0 + l;
    #pragma unroll
    for (int r = 0; r < 8; ++r) {
        const int m = r + 8 * half;
        ob[(size_t)m * HH] = acc[r];
    }
}

extern "C" void kernel_launch(void* const* d_in, const int* in_sizes, int n_in,
                              void* d_out, int out_size, void* d_ws, size_t ws_size,
                              hipStream_t stream) {
    (void)in_sizes; (void)n_in; (void)d_ws; (void)ws_size; (void)out_size;
    const float* q = (const float*)d_in[0];
    const float* k = (const float*)d_in[1];
    const float* v = (const float*)d_in[2];
    const float* w = (const float*)d_in[3];
    float* out = (float*)d_out;
    dim3 grid(BB * (QN / QT));   // 512 workgroups
    dim3 block(256);             // 8 wave32s
    addattn_kernel<<<grid, block, 0, stream>>>(q, k, v, w, out);
}